// CARAFE_88373247083227
// MI455X (gfx1250) — compile-verified
//
#include <hip/hip_runtime.h>

typedef float v2f __attribute__((ext_vector_type(2)));
typedef float v8f __attribute__((ext_vector_type(8)));

#define B_   4
#define C_   256
#define CC_  64
#define H_   64
#define W_   64
#define HW_  4096
#define M2_  100   // SF^2 * 25 output mask channels

// -------------------------------------------------------------------------
// Kernel 1: 1x1 conv (channel compression) as f32 WMMA GEMM.
// xc[b][cc][pix] = sum_c w1[cc][c] * x[b][c][pix] + b1[cc]
// One wave per (b, 16-pixel tile). M=64 -> 4 tiles, K=256 -> 64 wmma steps.
// -------------------------------------------------------------------------
__global__ __launch_bounds__(128) void carafe_conv1(
    const float* __restrict__ x, const float* __restrict__ w1,
    const float* __restrict__ b1, float* __restrict__ xc)
{
  const int lane  = threadIdx.x & 31;
  const int wid   = blockIdx.x * (blockDim.x >> 5) + (threadIdx.x >> 5);
  const int ntile = wid & 255;          // 256 pixel tiles
  const int b     = wid >> 8;           // batch
  const int half  = lane >> 4;          // K-half select
  const int l16   = lane & 15;
  const int pix   = ntile * 16 + l16;

  v8f acc[4] = {};
  const float* xb = x + (size_t)b * C_ * HW_;

  for (int kb = 0; kb < C_; kb += 4) {
    const int kc = kb + 2 * half;
    // B fragment: B[k][n] = x[b][kc+v][pix]
    v2f bfrag;
    const float* bp = xb + (size_t)kc * HW_ + pix;
    bfrag.x = bp[0];
    bfrag.y = bp[HW_];
#pragma unroll
    for (int mt = 0; mt < 4; ++mt) {
      // A fragment: A[m][k] = w1[mt*16+l16][kc+v]
      const float* ap = w1 + (size_t)(mt * 16 + l16) * C_ + kc;
      v2f afrag;
      afrag.x = ap[0];
      afrag.y = ap[1];
      acc[mt] = __builtin_amdgcn_wmma_f32_16x16x4_f32(
          false, afrag, false, bfrag, (short)0, acc[mt], false, false);
    }
  }

#pragma unroll
  for (int mt = 0; mt < 4; ++mt)
#pragma unroll
    for (int r = 0; r < 8; ++r) {
      const int m = mt * 16 + r + 8 * half;      // D layout: VGPR r -> M=r / r+8
      xc[((size_t)b * CC_ + m) * HW_ + pix] = acc[mt][r] + b1[m];
    }
}

// -------------------------------------------------------------------------
// Kernel 2: 3x3 conv -> 100 mask logit channels, as 9 shifted K=64 WMMA GEMMs.
// M = 100 padded to 112 (7 tiles of 16). Stores biased logits (pre-softmax).
// -------------------------------------------------------------------------
__global__ __launch_bounds__(128) void carafe_conv2(
    const float* __restrict__ xc, const float* __restrict__ w2,
    const float* __restrict__ b2, float* __restrict__ logits)
{
  const int lane  = threadIdx.x & 31;
  const int wid   = blockIdx.x * (blockDim.x >> 5) + (threadIdx.x >> 5);
  const int ntile = wid & 255;
  const int b     = wid >> 8;
  const int half  = lane >> 4;
  const int l16   = lane & 15;
  const int h     = ntile >> 2;              // pixel row (16 pixels share it)
  const int wb    = (ntile & 3) * 16;        // pixel col base

  v8f acc[7] = {};
  const float* xcb = xc + (size_t)b * CC_ * HW_;

  for (int dy = 0; dy < 3; ++dy) {
    const int  hh = h + dy - 1;
    const bool vh = (unsigned)hh < (unsigned)H_;
    for (int dx = 0; dx < 3; ++dx) {
      const int  ww  = wb + l16 + dx - 1;
      const bool ok  = vh && ((unsigned)ww < (unsigned)W_);
      const int  poff = ok ? (hh * W_ + ww) : 0;     // safe clamped offset
      const int  tap  = dy * 3 + dx;
      for (int kc4 = 0; kc4 < CC_; kc4 += 4) {
        const int c = kc4 + 2 * half;
        v2f bfrag;
        const float* bp = xcb + (size_t)c * HW_ + poff;
        bfrag.x = ok ? bp[0]   : 0.0f;
        bfrag.y = ok ? bp[HW_] : 0.0f;
#pragma unroll
        for (int mt = 0; mt < 7; ++mt) {
          const int m = mt * 16 + l16;
          v2f afrag;
          if (mt < 6) {
            // w2 is (100, 64, 3, 3): idx = m*576 + c*9 + tap
            const float* ap = w2 + (size_t)m * 576 + c * 9 + tap;
            afrag.x = ap[0];
            afrag.y = ap[9];                 // c+1
          } else {
            const bool  mok = l16 < 4;       // rows 96..99 valid, 100..111 pad
            const size_t idx = mok ? ((size_t)m * 576 + c * 9 + tap) : 0;
            afrag.x = mok ? w2[idx]     : 0.0f;
            afrag.y = mok ? w2[idx + 9] : 0.0f;
          }
          acc[mt] = __builtin_amdgcn_wmma_f32_16x16x4_f32(
              false, afrag, false, bfrag, (short)0, acc[mt], false, false);
        }
      }
    }
  }

  const int pix = ntile * 16 + l16;
#pragma unroll
  for (int mt = 0; mt < 7; ++mt)
#pragma unroll
    for (int r = 0; r < 8; ++r) {
      const int m = mt * 16 + r + 8 * half;
      if (m < M2_)
        logits[((size_t)b * M2_ + m) * HW_ + pix] = acc[mt][r] + b2[m];
    }
}

// -------------------------------------------------------------------------
// Kernel 3: in-register softmax over 25 taps (per quadrant) + reassembly.
// Thread = (pixel, 64-channel chunk). OOB taps folded into the mask (zeroed)
// with clamped addresses -> branch-free inner loop.
// Quadrants paired into v2f so the 100 FMAs/channel lower to 50 v_pk_fma_f32.
// -------------------------------------------------------------------------
__global__ __launch_bounds__(256) void carafe_upsample(
    const float* __restrict__ x, const float* __restrict__ logits,
    float* __restrict__ out)
{
  const int w   = threadIdx.x & 63;
  const int cg  = threadIdx.x >> 6;      // 4 channel groups of 64
  const int h   = blockIdx.x;
  const int b   = blockIdx.y;
  const int pix = h * 64 + w;

  // Load logits (coalesced over w) and softmax per quadrant.
  float mq[4][25];
  const float* lg = logits + (size_t)b * M2_ * HW_ + pix;
#pragma unroll
  for (int q = 0; q < 4; ++q) {
    float mx = -1e30f;
#pragma unroll
    for (int k = 0; k < 25; ++k) {
      const float v = lg[(size_t)(q * 25 + k) * HW_];
      mq[q][k] = v;
      mx = fmaxf(mx, v);
    }
    float s = 0.0f;
#pragma unroll
    for (int k = 0; k < 25; ++k) {
      const float e = __expf(mq[q][k] - mx);
      mq[q][k] = e;
      s += e;
    }
    const float inv = 1.0f / s;
#pragma unroll
    for (int k = 0; k < 25; ++k) mq[q][k] *= inv;
  }

  // Clamped tap offsets; fold zero-padding into the (already softmaxed) mask,
  // and pack quadrant pairs (q0,q1)/(q2,q3) for V_PK_FMA_F32.
  int off[25];
  v2f m01[25], m23[25];
#pragma unroll
  for (int ky = 0; ky < 5; ++ky)
#pragma unroll
    for (int kx = 0; kx < 5; ++kx) {
      const int  k  = ky * 5 + kx;
      const int  hh = h + ky - 2, ww = w + kx - 2;
      const bool okv = ((unsigned)hh < 64u) && ((unsigned)ww < 64u);
      const int  hc = min(max(hh, 0), 63), wc = min(max(ww, 0), 63);
      off[k] = hc * 64 + wc;
      const float f = okv ? 1.0f : 0.0f;
      m01[k].x = mq[0][k] * f;
      m01[k].y = mq[1][k] * f;
      m23[k].x = mq[2][k] * f;
      m23[k].y = mq[3][k] * f;
    }

  const float* xb = x + ((size_t)b * C_ + cg * 64) * HW_;
  float* ob = out + ((size_t)b * C_ + cg * 64) * (size_t)(128 * 128);

  for (int c = 0; c < 64; ++c) {
    const float* xcp = xb + (size_t)c * HW_;
    v2f a01 = {0.0f, 0.0f};   // (q0, q1) -> row 2h,   cols 2w, 2w+1
    v2f a23 = {0.0f, 0.0f};   // (q2, q3) -> row 2h+1, cols 2w, 2w+1
#pragma unroll
    for (int k = 0; k < 25; ++k) {
      const float v = xcp[off[k]];
      const v2f vv = {v, v};
      a01 = __builtin_elementwise_fma(vv, m01[k], a01);
      a23 = __builtin_elementwise_fma(vv, m23[k], a23);
    }
    float* op = ob + (size_t)c * (128 * 128) + (2 * h) * 128 + 2 * w;
    *(v2f*)(op)       = a01;
    *(v2f*)(op + 128) = a23;
  }
}

// -------------------------------------------------------------------------
extern "C" void kernel_launch(void* const* d_in, const int* in_sizes, int n_in,
                              void* d_out, int out_size, void* d_ws, size_t ws_size,
                              hipStream_t stream) {
  (void)in_sizes; (void)n_in; (void)out_size; (void)ws_size;
  const float* x  = (const float*)d_in[0];
  const float* w1 = (const float*)d_in[1];
  const float* b1 = (const float*)d_in[2];
  const float* w2 = (const float*)d_in[3];
  const float* b2 = (const float*)d_in[4];
  float* out = (float*)d_out;

  // Workspace: xc (4 MB) then logits (6.4 MB)
  float* xc     = (float*)d_ws;
  float* logits = xc + (size_t)B_ * CC_ * HW_;

  // 1024 waves each for conv1/conv2 (one wave per (batch, 16-pixel tile))
  carafe_conv1<<<256, 128, 0, stream>>>(x, w1, b1, xc);
  carafe_conv2<<<256, 128, 0, stream>>>(xc, w2, b2, logits);
  carafe_upsample<<<dim3(64, B_), 256, 0, stream>>>(x, logits, out);
}